// SiluAndMul_78752520339576
// MI455X (gfx1250) — compile-verified
//
#include <hip/hip_runtime.h>

// SwiGLU: out[r, c] = silu(x[r, c]) * x[r, c + D],  x: (4096, 2*D), D = 14336.
// HBM-streaming kernel: 128-bit non-temporal loads/stores, fp32 math with
// hardware rcp instead of the IEEE divide expansion (keeps VALU issue rate
// far below the rate needed to saturate 23.3 TB/s).

typedef __attribute__((ext_vector_type(4))) float v4f;

namespace {
constexpr int ROWS      = 4096;
constexpr int D         = 14336;      // output columns per row
constexpr int IN_STRIDE = 2 * D;      // 28672
constexpr int VEC       = 4;
constexpr int D4        = D / VEC;    // 3584 float4 columns per row
constexpr int BLOCK     = 256;        // 8 wave32 waves per block
constexpr int GRID_X    = D4 / BLOCK; // 14 blocks cover one row exactly
}

__global__ __launch_bounds__(BLOCK) void silu_mul_stream(
    const float* __restrict__ x, float* __restrict__ out) {
  const int row = blockIdx.y;
  const int c4  = blockIdx.x * BLOCK + threadIdx.x;   // float4 column, 0..D4-1

  const size_t in_base = (size_t)row * IN_STRIDE;
  const v4f* gate = reinterpret_cast<const v4f*>(x + in_base);
  const v4f* up   = reinterpret_cast<const v4f*>(x + in_base + D);
  v4f*       o    = reinterpret_cast<v4f*>(out + (size_t)row * D);

  // Streamed once: non-temporal 128-bit accesses (global_load_b128 th:NT).
  v4f g = __builtin_nontemporal_load(gate + c4);
  v4f u = __builtin_nontemporal_load(up + c4);

  v4f r;
#pragma unroll
  for (int i = 0; i < VEC; ++i) {
    const float gi = g[i];
    // silu(g) = g * sigmoid(g) = g * rcp(1 + exp(-g)).
    // __expf -> v_exp_f32 (trans), __builtin_amdgcn_rcpf -> v_rcp_f32 (trans):
    // ~6 VALU ops/element vs ~13 with the IEEE divide expansion.
    const float e = __expf(-gi);
    const float s = gi * __builtin_amdgcn_rcpf(1.0f + e);
    r[i] = s * u[i];
  }

  __builtin_nontemporal_store(r, o + c4);
}

extern "C" void kernel_launch(void* const* d_in, const int* in_sizes, int n_in,
                              void* d_out, int out_size, void* d_ws, size_t ws_size,
                              hipStream_t stream) {
  (void)in_sizes; (void)n_in; (void)out_size; (void)d_ws; (void)ws_size;
  const float* x   = reinterpret_cast<const float*>(d_in[0]);
  float*       out = reinterpret_cast<float*>(d_out);

  dim3 block(BLOCK);
  dim3 grid(GRID_X, ROWS);   // (14, 4096): exact tiling, fully coalesced
  silu_mul_stream<<<grid, block, 0, stream>>>(x, out);
}